// ModelResidualGIN_13932873908323
// MI455X (gfx1250) — compile-verified
//
#include <hip/hip_runtime.h>
#include <hip/hip_bf16.h>

// ---------------------------------------------------------------------------
// ResidualGIN forward for MI455X (gfx1250, wave32, WMMA).
//
// Shapes (compile-time constants from the reference):
//   N_NODES=100000, N_EDGES=1600000, N_FEAT=32, HIDDEN=64, N_LAYERS=4,
//   N_GRAPHS=512.  100000 = 6250*16  -> M tiles exactly divisible by 16.
// ---------------------------------------------------------------------------

#define N_NODES  100000
#define N_EDGES  1600000
#define N_FEAT   32
#define HIDDEN   64
#define N_LAYERS 4
#define N_GRAPHS 512
#define BN_EPS   1e-5f
#define M_TILES  (N_NODES / 16)   // 6250

typedef __attribute__((ext_vector_type(2))) float v2f;
typedef __attribute__((ext_vector_type(8))) float v8f;

// ---------------------------------------------------------------------------
// WMMA GEMM:  C[M x N] = opt_relu(A[M x K] @ W[K x N] + bias[N])
// 256 threads = 8 wave32 per block; each wave computes a 16 x N row tile
// with V_WMMA_F32_16X16X4_F32 (f32 in/out, K step 4).
//
// W is staged to LDS swizzled as lds[(k>>2)*N + n)*4 + (k&3)] so that the
// per-lane B fragment (two consecutive K for one column) is one ds_load_b64.
// A fragment per ISA layout: lane L holds row m0+(L&15), Ks khalf*2+{0,1}
// where khalf = L>>4.
// ---------------------------------------------------------------------------
template <int K, int N, bool RELU>
__global__ void __launch_bounds__(256)
gemm_wmma_kernel(const float* __restrict__ A,
                 const float* __restrict__ W,
                 const float* __restrict__ bias,
                 float* __restrict__ C)
{
    __shared__ float wlds[K * N];
    __shared__ float blds[N];

    // stage W (swizzled) + bias into LDS
    for (int i = threadIdx.x; i < K * N; i += 256) {
        int k = i / N;
        int n = i - k * N;
        wlds[((k >> 2) * N + n) * 4 + (k & 3)] = W[i];
    }
    if (threadIdx.x < N) blds[threadIdx.x] = bias[threadIdx.x];
    __syncthreads();

    const int wave  = threadIdx.x >> 5;
    const int lane  = threadIdx.x & 31;
    const int tile  = blockIdx.x * 8 + wave;
    if (tile >= M_TILES) return;            // wave-uniform: EXEC stays all-1s

    const int m0    = tile * 16;
    const int lrow  = lane & 15;
    const int khalf = lane >> 4;            // 0 or 1

    const float* arow = A + (size_t)(m0 + lrow) * K + khalf * 2;

    constexpr int NT = N / 16;
    v8f c[NT] = {};

#pragma unroll
    for (int k0 = 0; k0 < K; k0 += 4) {
        const v2f a = *(const v2f*)(arow + k0);          // global_load_b64
#pragma unroll
        for (int nt = 0; nt < NT; ++nt) {
            const int n = nt * 16 + lrow;
            const v2f b =
                *(const v2f*)&wlds[((k0 >> 2) * N + n) * 4 + khalf * 2]; // ds_load_b64
            c[nt] = __builtin_amdgcn_wmma_f32_16x16x4_f32(
                /*neg_a=*/false, a, /*neg_b=*/false, b,
                /*c_mod=*/(short)0, c[nt],
                /*reuse_a=*/false, /*reuse_b=*/false);
        }
    }

    // D layout: lane L, component r -> m = m0 + r + khalf*8, n = n0 + (L&15)
#pragma unroll
    for (int nt = 0; nt < NT; ++nt) {
        const int n  = nt * 16 + lrow;
        const float bv = blds[n];
#pragma unroll
        for (int r = 0; r < 8; ++r) {
            const int m = m0 + r + khalf * 8;
            float v = c[nt][r] + bv;
            if (RELU) v = v > 0.f ? v : 0.f;
            C[(size_t)m * N + n] = v;
        }
    }
}

// ---------------------------------------------------------------------------
// agg <- h  (seed the GIN self term, since agg = h + sum_neighbors),
// and zero the BN stats accumulator for this layer.
// ---------------------------------------------------------------------------
__global__ void copy_seed_kernel(const float4* __restrict__ h4,
                                 float4* __restrict__ agg4,
                                 float* __restrict__ stats, int n4)
{
    int tid = blockIdx.x * blockDim.x + threadIdx.x;
    if (tid < 2 * HIDDEN) stats[tid] = 0.f;
    if (tid < n4) agg4[tid] = h4[tid];
}

// ---------------------------------------------------------------------------
// Edge scatter: agg[dst] += h[src]. One thread per (edge, 4-float chunk).
// ---------------------------------------------------------------------------
__global__ void scatter_kernel(const float* __restrict__ h,
                               const int* __restrict__ src,
                               const int* __restrict__ dst,
                               float* __restrict__ agg)
{
    int tid = blockIdx.x * blockDim.x + threadIdx.x;
    int e = tid >> 4;
    if (e >= N_EDGES) return;
    int c = (tid & 15) * 4;
    int s = src[e];
    int d = dst[e];
    float4 v = *(const float4*)(h + (size_t)s * HIDDEN + c);
    float* o = agg + (size_t)d * HIDDEN + c;
    atomicAdd(o + 0, v.x);
    atomicAdd(o + 1, v.y);
    atomicAdd(o + 2, v.z);
    atomicAdd(o + 3, v.w);
}

// ---------------------------------------------------------------------------
// BN column statistics: sum and sumsq per feature (64 columns) over rows.
// 256 threads = 4 rows x 64 cols per step, LDS reduce, atomic flush.
// ---------------------------------------------------------------------------
__global__ void __launch_bounds__(256)
bn_stats_kernel(const float* __restrict__ u, float* __restrict__ stats)
{
    const int col  = threadIdx.x & 63;
    const int rofs = threadIdx.x >> 6;   // 0..3
    float s = 0.f, sq = 0.f;
    for (int r = blockIdx.x * 4 + rofs; r < N_NODES; r += gridDim.x * 4) {
        float v = u[(size_t)r * HIDDEN + col];
        s  += v;
        sq += v * v;
    }
    __shared__ float ls[2][4][HIDDEN];
    ls[0][rofs][col] = s;
    ls[1][rofs][col] = sq;
    __syncthreads();
    if (threadIdx.x < HIDDEN) {
        float ts = ls[0][0][col] + ls[0][1][col] + ls[0][2][col] + ls[0][3][col];
        float tq = ls[1][0][col] + ls[1][1][col] + ls[1][2][col] + ls[1][3][col];
        atomicAdd(&stats[col], ts);
        atomicAdd(&stats[HIDDEN + col], tq);
    }
}

// ---------------------------------------------------------------------------
// BN finalize: scale = gamma * rsqrt(var + eps); shift = beta - mean*scale.
// ---------------------------------------------------------------------------
__global__ void bn_finalize_kernel(const float* __restrict__ stats,
                                   const float* __restrict__ gamma,
                                   const float* __restrict__ beta,
                                   float* __restrict__ bnp)
{
    int c = threadIdx.x;
    if (c >= HIDDEN) return;
    const float invN = 1.0f / (float)N_NODES;
    float mean  = stats[c] * invN;
    float var   = stats[HIDDEN + c] * invN - mean * mean;
    float scale = gamma[c] * rsqrtf(var + BN_EPS);
    bnp[c]          = scale;
    bnp[HIDDEN + c] = beta[c] - mean * scale;
}

// ---------------------------------------------------------------------------
// h = relu(u*scale + shift) + h   (BN + ReLU + residual, elementwise)
// ---------------------------------------------------------------------------
__global__ void bn_apply_kernel(const float* __restrict__ u,
                                const float* __restrict__ bnp,
                                float* __restrict__ h, int total)
{
    int tid = blockIdx.x * blockDim.x + threadIdx.x;
    if (tid >= total) return;
    int c = tid & (HIDDEN - 1);
    float v = fmaf(u[tid], bnp[c], bnp[HIDDEN + c]);
    v = v > 0.f ? v : 0.f;
    h[tid] = h[tid] + v;
}

// ---------------------------------------------------------------------------
// Zero the pooled-graph buffer.
// ---------------------------------------------------------------------------
__global__ void zero_kernel(float* __restrict__ p, int n)
{
    int tid = blockIdx.x * blockDim.x + threadIdx.x;
    if (tid < n) p[tid] = 0.f;
}

// ---------------------------------------------------------------------------
// global_add_pool: g[batch[n]] += h[n]. One thread per (node, 4-float chunk).
// ---------------------------------------------------------------------------
__global__ void pool_kernel(const float* __restrict__ h,
                            const int* __restrict__ batch,
                            float* __restrict__ g)
{
    int tid = blockIdx.x * blockDim.x + threadIdx.x;
    int node = tid >> 4;
    if (node >= N_NODES) return;
    int c = (tid & 15) * 4;
    int b = batch[node];
    float4 v = *(const float4*)(h + (size_t)node * HIDDEN + c);
    float* o = g + (size_t)b * HIDDEN + c;
    atomicAdd(o + 0, v.x);
    atomicAdd(o + 1, v.y);
    atomicAdd(o + 2, v.z);
    atomicAdd(o + 3, v.w);
}

// ---------------------------------------------------------------------------
// Readout head: out[g] = relu(g@Wo1 + bo1) @ Wo2 + bo2.  One block per graph.
// ---------------------------------------------------------------------------
__global__ void __launch_bounds__(64)
head_kernel(const float* __restrict__ g,
            const float* __restrict__ Wo1, const float* __restrict__ bo1,
            const float* __restrict__ Wo2, const float* __restrict__ bo2,
            float* __restrict__ out)
{
    const int gid = blockIdx.x;
    const int t   = threadIdx.x;
    const float* row = g + (size_t)gid * HIDDEN;
    float acc = bo1[t];
#pragma unroll
    for (int k = 0; k < HIDDEN; ++k)
        acc = fmaf(row[k], Wo1[k * HIDDEN + t], acc);
    acc = acc > 0.f ? acc : 0.f;
    __shared__ float red[HIDDEN];
    red[t] = acc * Wo2[t];
    __syncthreads();
    for (int s = HIDDEN / 2; s > 0; s >>= 1) {
        if (t < s) red[t] += red[t + s];
        __syncthreads();
    }
    if (t == 0) out[gid] = red[0] + bo2[0];
}

// ---------------------------------------------------------------------------
// Launcher
// ---------------------------------------------------------------------------
extern "C" void kernel_launch(void* const* d_in, const int* in_sizes, int n_in,
                              void* d_out, int out_size, void* d_ws, size_t ws_size,
                              hipStream_t stream)
{
    (void)in_sizes; (void)n_in; (void)out_size; (void)ws_size;

    const float* x        = (const float*)d_in[0];
    const int*   ei       = (const int*)  d_in[1];   // [2, N_EDGES]
    /* edge_attr d_in[2] unused */
    const int*   batch    = (const int*)  d_in[3];
    const float* W_init   = (const float*)d_in[4];
    const float* b_init   = (const float*)d_in[5];
    const float* W1       = (const float*)d_in[6];   // [L,64,128]
    const float* b1       = (const float*)d_in[7];   // [L,128]
    const float* W2       = (const float*)d_in[8];   // [L,128,64]
    const float* b2       = (const float*)d_in[9];   // [L,64]
    const float* gamma    = (const float*)d_in[10];
    const float* beta     = (const float*)d_in[11];
    const float* Wo1      = (const float*)d_in[12];
    const float* bo1      = (const float*)d_in[13];
    const float* Wo2      = (const float*)d_in[14];
    const float* bo2      = (const float*)d_in[15];
    float*       out      = (float*)d_out;

    const int* src = ei;
    const int* dst = ei + N_EDGES;

    // workspace carve-up (bytes)
    char* ws = (char*)d_ws;
    const size_t szH = (size_t)N_NODES * HIDDEN * sizeof(float);      // 25.6 MB
    const size_t szT = (size_t)N_NODES * 2 * HIDDEN * sizeof(float);  // 51.2 MB
    float* d_h     = (float*)(ws);                         // [N,64]  h
    float* d_m     = (float*)(ws + szH);                   // [N,64]  agg/m then u
    float* d_t     = (float*)(ws + 2 * szH);               // [N,128] hidden of MLP
    float* d_stats = (float*)(ws + 2 * szH + szT);         // [128]
    float* d_bnp   = d_stats + 2 * HIDDEN;                 // [128]
    float* d_g     = d_bnp + 2 * HIDDEN;                   // [512,64]

    const int gemmBlocks    = (M_TILES + 7) / 8;                 // 782
    const int n4            = N_NODES * HIDDEN / 4;              // 1.6M float4
    const int copyBlocks    = (n4 + 255) / 256;
    const int scatterBlocks = (N_EDGES * 16 + 255) / 256;
    const int applyBlocks   = (N_NODES * HIDDEN + 255) / 256;
    const int poolBlocks    = (N_NODES * 16 + 255) / 256;

    // ---- initial projection: h = x @ W_init + b_init -----------------------
    gemm_wmma_kernel<N_FEAT, HIDDEN, false>
        <<<gemmBlocks, 256, 0, stream>>>(x, W_init, b_init, d_h);

    // ---- GIN layers --------------------------------------------------------
    for (int L = 0; L < N_LAYERS; ++L) {
        // m = h (self term), stats = 0
        copy_seed_kernel<<<copyBlocks, 256, 0, stream>>>(
            (const float4*)d_h, (float4*)d_m, d_stats, n4);
        // m += sum_{j in N(i)} h_j
        scatter_kernel<<<scatterBlocks, 256, 0, stream>>>(d_h, src, dst, d_m);
        // t = relu(m @ W1 + b1)
        gemm_wmma_kernel<HIDDEN, 2 * HIDDEN, true>
            <<<gemmBlocks, 256, 0, stream>>>(
                d_m, W1 + (size_t)L * HIDDEN * 2 * HIDDEN,
                b1 + (size_t)L * 2 * HIDDEN, d_t);
        // u = t @ W2 + b2   (overwrites m; m no longer needed)
        gemm_wmma_kernel<2 * HIDDEN, HIDDEN, false>
            <<<gemmBlocks, 256, 0, stream>>>(
                d_t, W2 + (size_t)L * 2 * HIDDEN * HIDDEN,
                b2 + (size_t)L * HIDDEN, d_m);
        // batch statistics
        bn_stats_kernel<<<256, 256, 0, stream>>>(d_m, d_stats);
        bn_finalize_kernel<<<1, HIDDEN, 0, stream>>>(
            d_stats, gamma + (size_t)L * HIDDEN, beta + (size_t)L * HIDDEN, d_bnp);
        // h = relu(BN(u)) + h
        bn_apply_kernel<<<applyBlocks, 256, 0, stream>>>(
            d_m, d_bnp, d_h, N_NODES * HIDDEN);
    }

    // ---- global_add_pool + head -------------------------------------------
    zero_kernel<<<(N_GRAPHS * HIDDEN + 255) / 256, 256, 0, stream>>>(
        d_g, N_GRAPHS * HIDDEN);
    pool_kernel<<<poolBlocks, 256, 0, stream>>>(d_h, batch, d_g);
    head_kernel<<<N_GRAPHS, HIDDEN, 0, stream>>>(d_g, Wo1, bo1, Wo2, bo2, out);
}